// Stag_62508954026129
// MI455X (gfx1250) — compile-verified
//
#include <hip/hip_runtime.h>
#include <hip/hip_bf16.h>

// Problem constants (match reference).
#define N_NODES 100000
#define N_EDGES 1600000
#define IN_F    128
#define OUT_F   64
#define SLOPE   0.2f

typedef __attribute__((ext_vector_type(2))) float v2f;
typedef __attribute__((ext_vector_type(8))) float v8f;

// ---------- monotone float <-> uint ordering for atomicMax-based segment max ----------
__device__ __forceinline__ unsigned f2ord(float f) {
  unsigned u = __float_as_uint(f);
  return (u & 0x80000000u) ? ~u : (u | 0x80000000u);
}
__device__ __forceinline__ float ord2f(unsigned u) {
  return __uint_as_float((u & 0x80000000u) ? (u & 0x7FFFFFFFu) : ~u);
}

// ---------- init: out = bias (folds final +bias); emax = -inf sentinel; esum = 0 ----------
__global__ void __launch_bounds__(256)
init_out_kernel(float* __restrict__ out, const float* __restrict__ bias,
                unsigned* __restrict__ emax_u, float* __restrict__ esum) {
  int i = blockIdx.x * blockDim.x + threadIdx.x;
  if (i >= N_NODES * OUT_F) return;
  out[i] = bias[i & (OUT_F - 1)];
  if ((i & (OUT_F - 1)) == 0) {
    emax_u[i >> 6] = 0u;   // 0 is the minimum of the ordered-uint mapping
    esum[i >> 6]   = 0.0f;
  }
}

// ---------- h = feat @ W_fc via V_WMMA_F32_16X16X4_F32 ----------
// One wave per 16x16 output tile; 4 waves per block cover the 64 output cols.
// fp32 A (16x4) layout: lanes 0-15 -> K = k0,k0+1 ; lanes 16-31 -> K = k0+2,k0+3 (M = lane%16).
// fp32 B (4x16) layout: symmetric (N = lane%16).
__global__ void __launch_bounds__(128)
gemm_fc_wmma(const float* __restrict__ feat, const float* __restrict__ Wfc,
             float* __restrict__ h) {
  const int lane = threadIdx.x & 31;
  const int wave = threadIdx.x >> 5;          // 0..3 : column tile
  const int row0 = blockIdx.x * 16;
  const int col0 = wave * 16;
  const int hi   = lane >> 4;                 // half-wave select
  const int lr   = lane & 15;
  const int row  = row0 + lr;                 // M index carried by this lane (A side)
  const int col  = col0 + lr;                 // N index carried by this lane (B/C side)

  const float* arow = feat + (size_t)row * IN_F + 2 * hi;
  const float* bcol = Wfc  + (size_t)(2 * hi) * OUT_F + col;

  v8f acc = {};
  #pragma unroll 8
  for (int k0 = 0; k0 < IN_F; k0 += 4) {
    v2f a = *(const v2f*)(arow + k0);         // K = k0+2*hi, k0+2*hi+1
    v2f b;
    b.x = bcol[(size_t)k0 * OUT_F];           // K = k0+2*hi   row of W_fc
    b.y = bcol[(size_t)k0 * OUT_F + OUT_F];   // K = k0+2*hi+1 row of W_fc
    acc = __builtin_amdgcn_wmma_f32_16x16x4_f32(
        /*neg_a=*/false, a, /*neg_b=*/false, b,
        /*c_mod=*/(short)0, acc, /*reuse_a=*/false, /*reuse_b=*/false);
  }

  // C/D layout: VGPR j -> M = j + 8*hi, N = lane%16
  float* hrow = h + (size_t)(row0 + 8 * hi) * OUT_F + col;
  #pragma unroll
  for (int j = 0; j < 8; ++j)
    hrow[(size_t)j * OUT_F] = acc[j];
}

// ---------- pp = h @ W_post + b_post ; columns -> loc_l, loc_r, lsl, lsr ----------
__global__ void __launch_bounds__(256)
post_proj_kernel(const float* __restrict__ h, const float* __restrict__ Wp,
                 const float* __restrict__ bp,
                 float* __restrict__ loc_l, float* __restrict__ loc_r,
                 float* __restrict__ lsl,  float* __restrict__ lsr) {
  int n = blockIdx.x * blockDim.x + threadIdx.x;
  if (n >= N_NODES) return;
  float4 acc = make_float4(bp[0], bp[1], bp[2], bp[3]);
  const float* hr = h + (size_t)n * OUT_F;
  #pragma unroll 8
  for (int k = 0; k < OUT_F; ++k) {
    const float hv = hr[k];
    const float4 w = *(const float4*)(Wp + 4 * k);  // W_post row k is a natural float4
    acc.x += hv * w.x; acc.y += hv * w.y; acc.z += hv * w.z; acc.w += hv * w.w;
  }
  loc_l[n] = acc.x; loc_r[n] = acc.y; lsl[n] = acc.z; lsr[n] = acc.w;
}

// ---------- per-edge logits + segment max ----------
__global__ void __launch_bounds__(256)
edge_logits_kernel(const int* __restrict__ src, const int* __restrict__ dst,
                   const float* __restrict__ loc_l, const float* __restrict__ loc_r,
                   const float* __restrict__ lsl,  const float* __restrict__ lsr,
                   const float* __restrict__ noise,
                   float* __restrict__ e, unsigned* __restrict__ emax_u) {
  int i = blockIdx.x * blockDim.x + threadIdx.x;
  if (i >= N_EDGES) return;
  const int s = src[i], d = dst[i];
  float le = loc_l[s] + loc_r[d];
  le = (le >= 0.0f) ? le : SLOPE * le;                 // leaky relu
  const float ev = le + __expf(lsl[s] + lsr[d]) * noise[i];
  e[i] = ev;
  atomicMax(emax_u + d, f2ord(ev));                    // GLOBAL_ATOMIC_MAX_U32 in L2
}

// ---------- exp(e - emax[dst]) in place + segment sum ----------
__global__ void __launch_bounds__(256)
edge_exp_kernel(const int* __restrict__ dst, float* __restrict__ e,
                const unsigned* __restrict__ emax_u, float* __restrict__ esum) {
  int i = blockIdx.x * blockDim.x + threadIdx.x;
  if (i >= N_EDGES) return;
  const int d = dst[i];
  const float ex = __expf(e[i] - ord2f(emax_u[d]));
  e[i] = ex;                                           // reuse e[] for e_exp
  atomicAdd(esum + d, ex);                             // GLOBAL_ATOMIC_ADD_F32 in L2
}

// ---------- a = e_exp/esum[dst]; out[dst] += h[src] * a  (32 lanes per edge) ----------
__global__ void __launch_bounds__(256)
edge_scatter_kernel(const int* __restrict__ src, const int* __restrict__ dst,
                    const float* __restrict__ e_exp, const float* __restrict__ esum,
                    const float* __restrict__ h, float* __restrict__ out) {
  const long long t = (long long)blockIdx.x * blockDim.x + threadIdx.x;
  const int i = (int)(t >> 5);          // edge
  const int j = (int)(t & 31);          // feature pair: handles 2j, 2j+1
  if (i >= N_EDGES) return;
  const int d = dst[i];
  const float a = e_exp[i] / esum[d];   // broadcast within the wave
  const int s = src[i];
  const v2f hv = *(const v2f*)(h + (size_t)s * OUT_F + 2 * j);   // coalesced 256B gather
  float* op = out + (size_t)d * OUT_F + 2 * j;
  atomicAdd(op,     hv.x * a);
  atomicAdd(op + 1, hv.y * a);
}

extern "C" void kernel_launch(void* const* d_in, const int* in_sizes, int n_in,
                              void* d_out, int out_size, void* d_ws, size_t ws_size,
                              hipStream_t stream) {
  const float* feat  = (const float*)d_in[0];
  const float* Wfc   = (const float*)d_in[1];
  const float* Wp    = (const float*)d_in[2];
  const float* bp    = (const float*)d_in[3];
  const float* bias  = (const float*)d_in[4];
  const float* noise = (const float*)d_in[5];
  const int*   src   = (const int*)d_in[6];
  const int*   dst   = (const int*)d_in[7];
  float* out = (float*)d_out;

  // Workspace carve-up (~34.5 MB)
  char* p = (char*)d_ws;
  float*    h      = (float*)p;    p += (size_t)N_NODES * OUT_F * sizeof(float);
  float*    loc_l  = (float*)p;    p += (size_t)N_NODES * sizeof(float);
  float*    loc_r  = (float*)p;    p += (size_t)N_NODES * sizeof(float);
  float*    lsl    = (float*)p;    p += (size_t)N_NODES * sizeof(float);
  float*    lsr    = (float*)p;    p += (size_t)N_NODES * sizeof(float);
  unsigned* emax_u = (unsigned*)p; p += (size_t)N_NODES * sizeof(unsigned);
  float*    esum   = (float*)p;    p += (size_t)N_NODES * sizeof(float);
  float*    e      = (float*)p;    p += (size_t)N_EDGES * sizeof(float);

  // 1) out = bias; emax/esum init
  init_out_kernel<<<(N_NODES * OUT_F + 255) / 256, 256, 0, stream>>>(out, bias, emax_u, esum);
  // 2) h = feat @ W_fc  (fp32 WMMA)
  gemm_fc_wmma<<<N_NODES / 16, 128, 0, stream>>>(feat, Wfc, h);
  // 3) posterior params
  post_proj_kernel<<<(N_NODES + 255) / 256, 256, 0, stream>>>(h, Wp, bp, loc_l, loc_r, lsl, lsr);
  // 4) edge logits + segment max
  edge_logits_kernel<<<(N_EDGES + 255) / 256, 256, 0, stream>>>(src, dst, loc_l, loc_r, lsl, lsr,
                                                                noise, e, emax_u);
  // 5) exp + segment sum
  edge_exp_kernel<<<(N_EDGES + 255) / 256, 256, 0, stream>>>(dst, e, emax_u, esum);
  // 6) normalize + weighted scatter of h[src]
  const long long scatter_threads = (long long)N_EDGES * 32;
  edge_scatter_kernel<<<(unsigned)((scatter_threads + 255) / 256), 256, 0, stream>>>(
      src, dst, e, esum, h, out);
}